// loss_func_83691732730037
// MI455X (gfx1250) — compile-verified
//
#include <hip/hip_runtime.h>
#include <math.h>

// Problem geometry (fixed by the reference).
#define HH 384
#define WW 512
#define HW (HH * WW)                    // 196608
#define BB 4
#define NPRED 12
#define TILE_PIX 1024                   // pixels per tile (256 threads x 4)
#define TILES_PER_IMG (HW / TILE_PIX)   // 192  (HW divisible by 1024)
#define NTILES (BB * TILES_PER_IMG)     // 768 tiles per step
#define GRIDX 256                       // blocks per step -> 3 tiles/block
#define NTHREADS 256
#define NARR 5                          // flow0, flow1, info0, info1, info2 (info3 provably unused: clip(x,0,0)==0)

#define LN2F 0.6931471805599453f
#define L2EF 1.4426950408889634f
#define MAXFLOW2 196608.0f              // MAX_FLOW^2 = H*W

typedef float v2f __attribute__((ext_vector_type(2)));
typedef float v8f __attribute__((ext_vector_type(8)));

// ---- native transcendental helpers (v_exp_f32 / v_log_f32 are base-2) ----
__device__ __forceinline__ float fexp2(float x) {
#if __has_builtin(__builtin_amdgcn_exp2f)
  return __builtin_amdgcn_exp2f(x);
#else
  return exp2f(x);
#endif
}
__device__ __forceinline__ float flog2(float x) {
#if __has_builtin(__builtin_amdgcn_logf)
  return __builtin_amdgcn_logf(x);   // log base 2
#else
  return log2f(x);
#endif
}

// Issue one async DMA: 16 bytes from global -> LDS (CDNA5 async path, tracked by ASYNCcnt).
__device__ __forceinline__ void async_ld_b128(uint32_t lds_off, uint64_t gaddr) {
  asm volatile("global_load_async_to_lds_b128 %0, %1, off"
               :: "v"(lds_off), "v"(gaddr) : "memory");
}

// Full 32-lane sum via the matrix pipe: V_WMMA_F32_16X16X4_F32 with B = ones.
// A-VGPR0 = acc (K=0 for lanes 0-15, K=2 for lanes 16-31), A-VGPR1 = 0, so
// D[m][n] = acc[m] + acc[m+16] for every n. Summing the 8 D VGPRs per lane gives
// sum(rowsum[0..7]) in lanes 0-15 and sum(rowsum[8..15]) in lanes 16-31; one
// shfl_xor(16) completes the reduction (result broadcast to all lanes).
__device__ __forceinline__ float wave_sum32(float v) {
#if __has_builtin(__builtin_amdgcn_wmma_f32_16x16x4_f32)
  v2f a; a[0] = v;    a[1] = 0.0f;
  v2f b; b[0] = 1.0f; b[1] = 1.0f;
  v8f c = {};
  c = __builtin_amdgcn_wmma_f32_16x16x4_f32(false, a, false, b, (short)0, c, false, false);
  float t = ((c[0] + c[1]) + (c[2] + c[3])) + ((c[4] + c[5]) + (c[6] + c[7]));
  t += __shfl_xor(t, 16, 32);
  return t;
#else
#pragma unroll
  for (int off = 16; off > 0; off >>= 1) v += __shfl_down(v, off, 32);
  return v;  // valid in lane 0 (sufficient: only lane 0 consumes it)
#endif
}

// ws layout (floats): [0] mask count; [1..12] per-step masked sums; [13..24] per-step counts.
__global__ void zero_ws_kernel(float* __restrict__ ws) {
  if (threadIdx.x < 25) ws[threadIdx.x] = 0.0f;
}

__global__ __launch_bounds__(NTHREADS)
void lap_loss_kernel(const float* __restrict__ flow_preds,
                     const float* __restrict__ info_preds,
                     const float* __restrict__ flow_gt,
                     const float* __restrict__ valid,
                     float* __restrict__ ws) {
  // Double-buffered staging: 2 x 5 arrays x 1024 floats = 40 KB LDS.
  __shared__ __align__(16) float sm[2][NARR][TILE_PIX];
  __shared__ float red[3][NTHREADS / 32];

  const int tid = threadIdx.x;
  const int s   = blockIdx.y;            // prediction step 0..11
  const int G   = gridDim.x;

  const uint32_t ldsA = (uint32_t)(uintptr_t)&sm[0][0][0];
  const uint32_t ldsB = (uint32_t)(uintptr_t)&sm[1][0][0];

  // DMA one tile's 5 channel slabs into an LDS buffer (each lane moves its own 16B per slab).
  auto prefetch = [&](int T, uint32_t lbase) {
    const int b   = T / TILES_PER_IMG;
    const int hw0 = (T % TILES_PER_IMG) * TILE_PIX;
    const uint64_t toff = (uint64_t)(tid * 16);
    const uint64_t fbase = (uint64_t)(uintptr_t)(flow_preds + ((size_t)(s * BB + b) * 2) * HW + hw0);
    const uint64_t ibase = (uint64_t)(uintptr_t)(info_preds + ((size_t)(s * BB + b) * 4) * HW + hw0);
    const uint64_t plane = (uint64_t)HW * 4u;   // bytes per channel plane
#pragma unroll
    for (int a = 0; a < 2; ++a)
      async_ld_b128(lbase + (uint32_t)(a * TILE_PIX * 4 + tid * 16), fbase + (uint64_t)a * plane + toff);
#pragma unroll
    for (int a = 0; a < 3; ++a)   // info channels 0,1,2 only (channel 3 mathematically unused)
      async_ld_b128(lbase + (uint32_t)((2 + a) * TILE_PIX * 4 + tid * 16), ibase + (uint64_t)a * plane + toff);
  };

  float acc_sum = 0.0f, acc_cnt = 0.0f, acc_msk = 0.0f;

  // Software pipeline prologue: two tiles in flight (10 async DMAs / wave).
  const int T0 = blockIdx.x;
  if (T0 < NTILES)     prefetch(T0, ldsA);
  if (T0 + G < NTILES) prefetch(T0 + G, ldsB);

  int bufsel = 0;
  for (int T = T0; T < NTILES; T += G, bufsel ^= 1) {
    // Wait for the current tile's 5 DMAs (next tile's 5 may remain outstanding).
    if (T + G < NTILES) asm volatile("s_wait_asynccnt 5" ::: "memory");
    else                asm volatile("s_wait_asynccnt 0" ::: "memory");

    const int b   = T / TILES_PER_IMG;
    const int hw0 = (T % TILES_PER_IMG) * TILE_PIX;

    // gt / valid stay L2-resident (reused by all 12 steps): plain b128 loads.
    float4 G0 = *(const float4*)(flow_gt + (size_t)(2 * b) * HW + hw0 + tid * 4);
    float4 G1 = *(const float4*)(flow_gt + (size_t)(2 * b + 1) * HW + hw0 + tid * 4);
    float4 VV = *(const float4*)(valid + (size_t)b * HW + hw0 + tid * 4);
    const float* g0v = (const float*)&G0;
    const float* g1v = (const float*)&G1;
    const float* vvv = (const float*)&VV;

    const float* f0v = &sm[bufsel][0][tid * 4];
    const float* f1v = &sm[bufsel][1][tid * 4];
    const float* a0v = &sm[bufsel][2][tid * 4];
    const float* a1v = &sm[bufsel][3][tid * 4];
    const float* l0v = &sm[bufsel][4][tid * 4];

#pragma unroll
    for (int k = 0; k < 4; ++k) {
      const float g0 = g0v[k], g1 = g1v[k];
      const float a0 = a0v[k], a1 = a1v[k];
      const float lb0   = fminf(fmaxf(l0v[k], 0.0f), 10.0f);   // clip(info2, 0, 10); lb1 == 0
      const float ebinv = fexp2(-lb0 * L2EF);                  // exp(-lb0)

      // logsumexp(alpha) = max + ln(1 + exp(-|a0-a1|))
      const float lseA = fmaxf(a0, a1) +
                         flog2(1.0f + fexp2(-fabsf(a0 - a1) * L2EF)) * LN2F;

      const float mag2   = g0 * g0 + g1 * g1;
      const bool  maskok = (mag2 < MAXFLOW2) && (vvv[k] != 0.0f);

      // c = 0
      {
        const float d  = fabsf(g0 - f0v[k]);
        const float x0 = a0 - LN2F - lb0 - d * ebinv;
        const float x1 = a1 - LN2F - d;
        const float lse = fmaxf(x0, x1) +
                          flog2(1.0f + fexp2(-fabsf(x0 - x1) * L2EF)) * LN2F;
        const float nf = lseA - lse;
        const bool ok = maskok && isfinite(nf);
        acc_sum += ok ? nf : 0.0f;
        acc_cnt += ok ? 1.0f : 0.0f;
      }
      // c = 1
      {
        const float d  = fabsf(g1 - f1v[k]);
        const float x0 = a0 - LN2F - lb0 - d * ebinv;
        const float x1 = a1 - LN2F - d;
        const float lse = fmaxf(x0, x1) +
                          flog2(1.0f + fexp2(-fabsf(x0 - x1) * L2EF)) * LN2F;
        const float nf = lseA - lse;
        const bool ok = maskok && isfinite(nf);
        acc_sum += ok ? nf : 0.0f;
        acc_cnt += ok ? 1.0f : 0.0f;
      }
      if (s == 0) acc_msk += maskok ? 1.0f : 0.0f;
    }

    // Reuse this LDS buffer for tile T+2G: our ds reads of it are consumed above.
    if (T + 2 * G < NTILES) {
      asm volatile("s_wait_dscnt 0" ::: "memory");
      prefetch(T + 2 * G, bufsel ? ldsB : ldsA);
    }
  }

  // ---- block reduction: WMMA 32-lane sum (matrix pipe), then cross-wave via LDS ----
  // No divergence reaches this point, so EXEC is all ones as WMMA requires.
  acc_sum = wave_sum32(acc_sum);
  acc_cnt = wave_sum32(acc_cnt);
  acc_msk = wave_sum32(acc_msk);

  const int wave = tid >> 5, lane = tid & 31;
  if (lane == 0) { red[0][wave] = acc_sum; red[1][wave] = acc_cnt; red[2][wave] = acc_msk; }
  __syncthreads();
  if (tid == 0) {
    float S = 0.0f, C = 0.0f, M = 0.0f;
#pragma unroll
    for (int w = 0; w < NTHREADS / 32; ++w) { S += red[0][w]; C += red[1][w]; M += red[2][w]; }
    atomicAdd(&ws[1 + s], S);
    atomicAdd(&ws[13 + s], C);
    if (s == 0) atomicAdd(&ws[0], M);
  }
}

__global__ void finalize_kernel(const float* __restrict__ ws, float* __restrict__ out) {
  float total = 0.0f;
#pragma unroll
  for (int s = 0; s < NPRED; ++s)
    total += ws[1 + s] / fmaxf(ws[13 + s], 1.0f);
  const float frac = ws[0] / (float)((size_t)BB * HW);
  out[0] = (frac < 0.25f) ? 0.0f : total;
}

extern "C" void kernel_launch(void* const* d_in, const int* in_sizes, int n_in,
                              void* d_out, int out_size, void* d_ws, size_t ws_size,
                              hipStream_t stream) {
  (void)in_sizes; (void)n_in; (void)out_size; (void)ws_size;
  const float* flow_preds = (const float*)d_in[0];
  const float* info_preds = (const float*)d_in[1];
  const float* flow_gt    = (const float*)d_in[2];
  const float* valid      = (const float*)d_in[3];
  // d_in[4] (auxiliary) contributes 0.0 * sum -> ignored.
  float* ws  = (float*)d_ws;
  float* out = (float*)d_out;

  zero_ws_kernel<<<1, 32, 0, stream>>>(ws);
  dim3 grid(GRIDX, NPRED, 1);
  lap_loss_kernel<<<grid, NTHREADS, 0, stream>>>(flow_preds, info_preds, flow_gt, valid, ws);
  finalize_kernel<<<1, 1, 0, stream>>>(ws, out);
}